// SimpleCRF_11630771438341
// MI455X (gfx1250) — compile-verified
//
#include <hip/hip_runtime.h>
#include <hip/hip_bf16.h>

// ---------------------------------------------------------------------------
// CRF forward on MI455X (gfx1250):
//  gold score    : two-stage deterministic reduction
//  partition fn  : scaled probability-space matrix scan
//                  p_{t} = p_{t-1} * E * diag(exp(e_t)),  E = exp(trans)
//                  chunk transfer matrices built with V_WMMA_F32_16X16X4_F32,
//                  emissions streamed into LDS by the Tensor Data Mover
//                  (tensor_load_to_lds + s_wait_tensorcnt), combined with a
//                  log2(NCHUNK) pairwise WMMA tree.
// ---------------------------------------------------------------------------

#define SEQ_LEN 262144
#define NT      128
#define NCHUNK  512
#define CHUNK_L ((SEQ_LEN - 1 + NCHUNK - 1) / NCHUNK)   // 512
#define MSTRIDE 132   // LDS row stride (132 % 64 == 4 -> conflict-free A reads)
#define SLAB_R  16    // emission rows per TDM slab

typedef __attribute__((ext_vector_type(2))) float v2f;
typedef __attribute__((ext_vector_type(8))) float v8f;
typedef __attribute__((ext_vector_type(4))) unsigned int v4u;
typedef __attribute__((ext_vector_type(8))) int v8i;
typedef __attribute__((ext_vector_type(4))) int v4i;

// ---------------------------------------------------------------- gold score
__global__ __launch_bounds__(256) void gold_partial_kernel(
    const float* __restrict__ emis, const int* __restrict__ tags,
    const float* __restrict__ trans, float* __restrict__ partials) {
  __shared__ float red[256];
  float acc = 0.f;
  const int stride = gridDim.x * blockDim.x;
  for (long s = blockIdx.x * blockDim.x + threadIdx.x; s < SEQ_LEN; s += stride) {
    const int t = tags[s];
    acc += emis[s * (long)NT + t];
    if (s + 1 < SEQ_LEN) acc += trans[t * NT + tags[s + 1]];
  }
  red[threadIdx.x] = acc;
  __syncthreads();
  for (int o = 128; o > 0; o >>= 1) {
    if (threadIdx.x < o) red[threadIdx.x] += red[threadIdx.x + o];
    __syncthreads();
  }
  if (threadIdx.x == 0) partials[blockIdx.x] = red[0];
}

__global__ __launch_bounds__(256) void gold_final_kernel(
    const float* __restrict__ partials, float* __restrict__ out) {
  __shared__ float red[256];
  red[threadIdx.x] = partials[threadIdx.x];
  __syncthreads();
  for (int o = 128; o > 0; o >>= 1) {
    if (threadIdx.x < o) red[threadIdx.x] += red[threadIdx.x + o];
    __syncthreads();
  }
  if (threadIdx.x == 0) out[0] = red[0];
}

// ------------------------------------------------------------------- E=exp(T)
__global__ __launch_bounds__(256) void expE_kernel(
    const float* __restrict__ trans, float* __restrict__ expE) {
  const int i = blockIdx.x * blockDim.x + threadIdx.x;
  if (i < NT * NT) expE[i] = __expf(trans[i]);
}

// -------------------------------------------------------- TDM slab load issue
// 2D descriptor: element = 4B, tile = [NT, SLAB_R] rows of the emission
// matrix starting at `grow`, destination = LDS byte address `lds_addr`.
__device__ __forceinline__ void tdm_issue_slab(const float* grow,
                                               unsigned remaining_rows,
                                               unsigned lds_addr) {
  const unsigned long long ga = (unsigned long long)(const void*)grow;
  // group0: count=1 | lds_addr | global_addr[56:0] | type=2
  v4u g0 = {1u, lds_addr, (unsigned)(ga & 0xFFFFFFFFull),
            (unsigned)((ga >> 32) & 0x01FFFFFFull) | (2u << 30)};
  // group1: data_size=4B, tensor_dim0=NT, tensor_dim1=remaining_rows,
  //         tile_dim0=NT, tile_dim1=SLAB_R, tensor_dim0_stride=NT
  const unsigned td1 = remaining_rows;
  v8i g1 = {(int)0x00020000u,                       // mask=0, data_size=2(4B)
            (int)((unsigned)NT << 16),              // abar=0 | td0[15:0]
            (int)(((td1 & 0xFFFFu) << 16) | 0u),    // td0[31:16]=0 | td1[15:0]
            (int)(((unsigned)NT << 16) | (td1 >> 16)), // td1[31:16] | tile0
            (int)SLAB_R,                            // tile1 | tile2=0
            (int)NT,                                // td0_stride[31:0]
            0, 0};                                  // td0_stride hi, td1_stride
  v4i gz4 = {0, 0, 0, 0};
  v8i gz8 = {0, 0, 0, 0, 0, 0, 0, 0};
  // 6-arg form (clang-23 / therock-10.0 headers)
  __builtin_amdgcn_tensor_load_to_lds(g0, g1, gz4, gz4, gz8, 0);
}

// --------------------------------------------------------- phase 1: chunk scan
// Each block builds T_c = Prod_{t in chunk} ( E * diag(exp(e_t)) ) in prob
// space, max-renormalized each step, log-scale accumulated in `lsum`.
// Invariant at exit: true_T = Mcur * exp(lsum).
__global__ __launch_bounds__(256) void chunk_kernel(
    const float* __restrict__ emis, const float* __restrict__ expE,
    float* __restrict__ mats, float* __restrict__ scales) {
  extern __shared__ float lds[];
  float* Ma   = lds;                       // NT * MSTRIDE
  float* Mb   = Ma + NT * MSTRIDE;         // NT * MSTRIDE
  float* wrow = Mb + NT * MSTRIDE;         // NT
  float* slab = wrow + NT;                 // 2 * SLAB_R * NT (TDM staging)
  __shared__ unsigned maxbits;

  const int tid  = threadIdx.x;
  const int wave = tid >> 5;
  const int lane = tid & 31;
  const int half = lane >> 4;              // 0: K pair {k,k+1}, 1: {k+2,k+3}
  const int l15  = lane & 15;

  const long base = 1 + (long)blockIdx.x * CHUNK_L;
  long endr = base + CHUNK_L;
  if (endr > SEQ_LEN) endr = SEQ_LEN;
  const int rows  = (int)(endr - base);
  const int nslab = (rows + SLAB_R - 1) / SLAB_R;

  // LDS byte address of slab buffer 0 (dynamic LDS starts after static LDS)
  const unsigned slab_lds0 =
      (unsigned)__builtin_amdgcn_groupstaticsize() +
      (unsigned)((const char*)slab - (const char*)lds);

  if (wave == 0)   // prefetch slab 0 via the Tensor Data Mover
    tdm_issue_slab(emis + base * NT, (unsigned)(SEQ_LEN - base), slab_lds0);
  if (tid == 0) maxbits = 0u;

  float* Mcur = Ma;
  float* Mnxt = Mb;
  float lsum = 0.f;

  for (int r = 0; r < rows; ++r) {
    if ((r & (SLAB_R - 1)) == 0) {
      const int si = r / SLAB_R;
      if (wave == 0) {
        __builtin_amdgcn_s_wait_tensorcnt(0);   // slab si landed in LDS
        if (si + 1 < nslab) {                   // overlap next slab with compute
          const long nr = base + (long)(si + 1) * SLAB_R;
          tdm_issue_slab(emis + nr * NT, (unsigned)(SEQ_LEN - nr),
                         slab_lds0 + (unsigned)(((si + 1) & 1) *
                                                SLAB_R * NT * sizeof(float)));
        }
      }
      __syncthreads();                          // slab visible to all waves
    }
    const float* erow =
        slab + ((((r / SLAB_R) & 1) * SLAB_R) + (r & (SLAB_R - 1))) * NT;

    if (r == 0) {
      // ---- init: Mcur = E * diag(exp(e_base))
      if (tid < NT) wrow[tid] = __expf(erow[tid]);
      __syncthreads();
      float lmax = 0.f;
      for (int idx = tid; idx < NT * NT; idx += 256) {
        const int rr = idx >> 7, cc = idx & (NT - 1);
        const float v = expE[idx] * wrow[cc];
        Ma[rr * MSTRIDE + cc] = v;
        lmax = fmaxf(lmax, v);
      }
      atomicMax(&maxbits, __float_as_uint(lmax));
      __syncthreads();
      continue;
    }

    const float prevmax = __uint_as_float(maxbits);
    lsum += __logf(prevmax);
    __syncthreads();
    if (tid == 0) maxbits = 0u;
    if (tid < NT) wrow[tid] = __expf(erow[tid]) * (1.0f / prevmax);
    __syncthreads();

    // ---- Mnxt = (Mcur x E) * diag(wrow), WMMA f32 16x16x4
    const int rb  = wave;                  // this wave's 16-row output block
    const int row = rb * 16 + l15;         // A-fragment row (lanes 0..15)
    v8f acc[8];
    const v8f zero = {0.f, 0.f, 0.f, 0.f, 0.f, 0.f, 0.f, 0.f};
#pragma unroll
    for (int cb = 0; cb < 8; ++cb) acc[cb] = zero;

    for (int k = 0; k < NT; k += 4) {
      const int ks = k + half * 2;
      v2f a;
      a.x = Mcur[row * MSTRIDE + ks];
      a.y = Mcur[row * MSTRIDE + ks + 1];
#pragma unroll
      for (int cb = 0; cb < 8; ++cb) {
        const int col = cb * 16 + l15;
        v2f b;
        b.x = expE[ks * NT + col];
        b.y = expE[(ks + 1) * NT + col];
        acc[cb] = __builtin_amdgcn_wmma_f32_16x16x4_f32(
            false, a, false, b, (short)0, acc[cb], false, false);
      }
    }

    float lmax = 0.f;
#pragma unroll
    for (int cb = 0; cb < 8; ++cb) {
      const int col = cb * 16 + l15;
      const float wc = wrow[col];
#pragma unroll
      for (int v = 0; v < 8; ++v) {
        const float val = acc[cb][v] * wc;
        Mnxt[(rb * 16 + v + half * 8) * MSTRIDE + col] = val;
        lmax = fmaxf(lmax, val);
      }
    }
    atomicMax(&maxbits, __float_as_uint(lmax));
    __syncthreads();
    float* tmp = Mcur; Mcur = Mnxt; Mnxt = tmp;
  }

  // ---- write chunk matrix (raw) + log-scale
  float* dst = mats + (long)blockIdx.x * NT * NT;
  for (int idx = tid; idx < NT * NT; idx += 256)
    dst[idx] = Mcur[(idx >> 7) * MSTRIDE + (idx & (NT - 1))];
  if (tid == 0) scales[blockIdx.x] = lsum;
}

// ------------------------------------------------- phase 2: pairwise combine
// out[p] = normalize( in[2p] x in[2p+1] ),  scale_out = s_a + s_b + log(max)
__global__ __launch_bounds__(256) void combine_kernel(
    const float* __restrict__ inM, const float* __restrict__ inS,
    float* __restrict__ outM, float* __restrict__ outS) {
  __shared__ unsigned maxbits;
  const int tid  = threadIdx.x;
  const int wave = tid >> 5;
  const int lane = tid & 31;
  const int half = lane >> 4;
  const int l15  = lane & 15;

  const float* A = inM + (long)(2 * blockIdx.x) * NT * NT;
  const float* B = inM + (long)(2 * blockIdx.x + 1) * NT * NT;
  if (tid == 0) maxbits = 0u;
  __syncthreads();

  const int rb  = wave;
  const int row = rb * 16 + l15;
  v8f acc[8];
  const v8f zero = {0.f, 0.f, 0.f, 0.f, 0.f, 0.f, 0.f, 0.f};
#pragma unroll
  for (int cb = 0; cb < 8; ++cb) acc[cb] = zero;

  for (int k = 0; k < NT; k += 4) {
    const int ks = k + half * 2;
    v2f a;
    a.x = A[row * NT + ks];
    a.y = A[row * NT + ks + 1];
#pragma unroll
    for (int cb = 0; cb < 8; ++cb) {
      const int col = cb * 16 + l15;
      v2f b;
      b.x = B[ks * NT + col];
      b.y = B[(ks + 1) * NT + col];
      acc[cb] = __builtin_amdgcn_wmma_f32_16x16x4_f32(
          false, a, false, b, (short)0, acc[cb], false, false);
    }
  }

  float lmax = 0.f;
#pragma unroll
  for (int cb = 0; cb < 8; ++cb)
#pragma unroll
    for (int v = 0; v < 8; ++v) lmax = fmaxf(lmax, acc[cb][v]);
  atomicMax(&maxbits, __float_as_uint(lmax));
  __syncthreads();

  const float m   = __uint_as_float(maxbits);
  const float inv = 1.0f / m;
  float* D = outM + (long)blockIdx.x * NT * NT;
#pragma unroll
  for (int cb = 0; cb < 8; ++cb) {
    const int col = cb * 16 + l15;
#pragma unroll
    for (int v = 0; v < 8; ++v)
      D[(rb * 16 + v + half * 8) * NT + col] = acc[cb][v] * inv;
  }
  if (tid == 0)
    outS[blockIdx.x] = inS[2 * blockIdx.x] + inS[2 * blockIdx.x + 1] + __logf(m);
}

// ---------------------------------------------------------- phase 3: finalize
__global__ __launch_bounds__(128) void finalize_kernel(
    const float* __restrict__ emis, const float* __restrict__ Mfin,
    const float* __restrict__ sFin, float* __restrict__ out) {
  __shared__ float p0[NT];
  __shared__ float red[NT];
  const int tid = threadIdx.x;

  red[tid] = emis[tid];
  __syncthreads();
  for (int o = 64; o > 0; o >>= 1) {
    if (tid < o) red[tid] = fmaxf(red[tid], red[tid + o]);
    __syncthreads();
  }
  const float m0 = red[0];
  __syncthreads();
  p0[tid] = __expf(emis[tid] - m0);
  __syncthreads();

  float v = 0.f;
  for (int i = 0; i < NT; ++i) v += p0[i] * Mfin[i * NT + tid];
  red[tid] = v;
  __syncthreads();
  for (int o = 64; o > 0; o >>= 1) {
    if (tid < o) red[tid] += red[tid + o];
    __syncthreads();
  }
  if (tid == 0) out[1] = m0 + __logf(red[0]) + sFin[0];
}

// ---------------------------------------------------------------------------
extern "C" void kernel_launch(void* const* d_in, const int* in_sizes, int n_in,
                              void* d_out, int out_size, void* d_ws, size_t ws_size,
                              hipStream_t stream) {
  const float* emis  = (const float*)d_in[0];
  const int*   tags  = (const int*)d_in[1];
  const float* trans = (const float*)d_in[2];
  float* out = (float*)d_out;

  // workspace layout (floats)
  float* ws        = (float*)d_ws;
  float* gold_part = ws;                                    // 256
  float* expE      = gold_part + 256;                       // 16384
  float* scalesA   = expE + NT * NT;                        // NCHUNK
  float* scalesB   = scalesA + NCHUNK;                      // NCHUNK/2
  float* matsA     = scalesB + NCHUNK;                      // NCHUNK   * 16384
  float* matsB     = matsA + (long)NCHUNK * NT * NT;        // NCHUNK/2 * 16384

  // gold score
  gold_partial_kernel<<<256, 256, 0, stream>>>(emis, tags, trans, gold_part);
  gold_final_kernel<<<1, 256, 0, stream>>>(gold_part, out);

  // E = exp(trans)
  expE_kernel<<<(NT * NT + 255) / 256, 256, 0, stream>>>(trans, expE);

  // phase 1: per-chunk transfer matrices (double-buffered LDS + TDM staging)
  const size_t lds_bytes =
      (size_t)(2 * NT * MSTRIDE + NT + 2 * SLAB_R * NT) * sizeof(float);
  chunk_kernel<<<NCHUNK, 256, lds_bytes, stream>>>(emis, expE, matsA, scalesA);

  // phase 2: log2(NCHUNK) combine rounds, ping-pong A<->B
  int count = NCHUNK;
  float* inM = matsA;  float* inS = scalesA;
  float* oM  = matsB;  float* oS  = scalesB;
  while (count > 1) {
    const int pairs = count / 2;
    combine_kernel<<<pairs, 256, 0, stream>>>(inM, inS, oM, oS);
    float* tM = inM; inM = oM; oM = tM;
    float* tS = inS; inS = oS; oS = tS;
    count = pairs;
  }

  // phase 3: p0 x T_total -> logsumexp
  finalize_kernel<<<1, NT, 0, stream>>>(emis, inM, inS, out);
}